// BaseNeighborlist_48241072668759
// MI455X (gfx1250) — compile-verified
//
#include <hip/hip_runtime.h>
#include <hip/hip_bf16.h>
#include <stdint.h>

// Problem constants (match reference)
#define NMOLS   32
#define NATOMS  512
#define NPAIRS  130816
#define CUT2    (5.2f * 5.2f)

#define BLOCK   256                 // 8 wave32 waves
#define ITERS   8
#define CHUNK   (BLOCK * ITERS)     // 2048 pairs per block
#define PB      ((NPAIRS + CHUNK - 1) / CHUNK)   // 64 blocks per molecule
#define NB      (NMOLS * PB)        // 2048 blocks total
#define TOTAL   ((long long)NMOLS * NPAIRS)      // 4,186,112 items

typedef __attribute__((ext_vector_type(4))) unsigned int u32x4;
typedef __attribute__((ext_vector_type(8))) int          i32x8;
typedef __attribute__((ext_vector_type(4))) int          i32x4;

#if defined(__has_builtin)
#  if __has_builtin(__builtin_amdgcn_tensor_load_to_lds)
#    define HAVE_TDM 1
#  endif
#endif
#ifndef HAVE_TDM
#  define HAVE_TDM 0
#endif

// Stage one molecule's coordinate table (512*3 f32 = 6KB) into LDS via the
// Tensor Data Mover. D# layout per CDNA5 ISA.
__device__ __forceinline__ void stage_coords(float* sC, const float* coords, int mol) {
#if HAVE_TDM
    if (threadIdx.x < 32) {   // wave 0 issues the DMA (uniform per-wave branch)
        uint64_t ga  = (uint64_t)(uintptr_t)(coords + (size_t)mol * (NATOMS * 3));
        uint32_t lds = (uint32_t)(uintptr_t)sC;  // low 32 bits == LDS byte offset
        // Group 0: count=1 | lds_addr | global_addr[56:0] | type=2
        u32x4 g0 = { 1u,
                     lds,
                     (uint32_t)ga,
                     (uint32_t)((ga >> 32) & 0x1FFFFFFull) | 0x80000000u };
        // Group 1: data_size=4B (code 2); tensor_dim0=1536; tensor_dim1=1;
        //          tile_dim0=1536; tensor_dim0_stride=1536; mask=0 (no cluster)
        i32x8 g1 = { 0x00020000,
                     (int)(1536u << 16),   // tensor_dim0[15:0] at bits 63:48
                     (int)(1u    << 16),   // tensor_dim1[15:0] at bits 95:80
                     (int)(1536u << 16),   // tile_dim0 at bits 127:112
                     0,                    // tile_dim1=0 (unused), tile_dim2=0
                     1536,                 // tensor_dim0_stride[31:0]
                     0, 0 };
        i32x4 z4 = { 0, 0, 0, 0 };
#if __clang_major__ >= 23
        i32x8 z8 = { 0, 0, 0, 0, 0, 0, 0, 0 };
        __builtin_amdgcn_tensor_load_to_lds(g0, g1, z4, z4, z8, 0);
#else
        __builtin_amdgcn_tensor_load_to_lds(g0, g1, z4, z4, 0);
#endif
        __builtin_amdgcn_s_wait_tensorcnt(0);
    }
#else
    const float* gp = coords + (size_t)mol * (NATOMS * 3);
    for (int i = threadIdx.x; i < NATOMS * 3; i += BLOCK) sC[i] = gp[i];
#endif
    __syncthreads();
}

// ---------------- Pass A: per-block valid counts ----------------
__global__ void nl_count_kernel(const float* __restrict__ coords,
                                const int*   __restrict__ nl,
                                const float* __restrict__ sv,
                                int* __restrict__ blkCnt) {
    __shared__ float sC[NATOMS * 3];
    __shared__ int   sW[8];
    const int mol = blockIdx.y;
    const int pb  = blockIdx.x;
    stage_coords(sC, coords, mol);

    const int lane = threadIdx.x & 31;
    const int wave = threadIdx.x >> 5;
    const int base = pb * CHUNK;

    int wcnt = 0;
    for (int it = 0; it < ITERS; ++it) {
        const int pair = base + it * BLOCK + threadIdx.x;
        bool valid = false;
        if (pair < NPAIRS) {
            const int a = nl[pair];
            const int b = nl[NPAIRS + pair];
            const float dx = sC[a * 3 + 0] - sC[b * 3 + 0] + sv[pair * 3 + 0];
            const float dy = sC[a * 3 + 1] - sC[b * 3 + 1] + sv[pair * 3 + 1];
            const float dz = sC[a * 3 + 2] - sC[b * 3 + 2] + sv[pair * 3 + 2];
            valid = (dx * dx + dy * dy + dz * dz) <= CUT2;
        }
        const unsigned mb = __builtin_amdgcn_ballot_w32(valid);
        wcnt += __popc(mb);
    }
    if (lane == 0) sW[wave] = wcnt;
    __syncthreads();
    if (threadIdx.x == 0) {
        int s = 0;
        for (int w = 0; w < 8; ++w) s += sW[w];
        blkCnt[mol * PB + pb] = s;
    }
}

// ---------------- Pass B: exclusive scan of block counts ----------------
__global__ void nl_scan_kernel(const int* __restrict__ cnt,
                               int* __restrict__ off, int nb) {
    __shared__ int s[256];
    const int t    = threadIdx.x;
    const int base = t * 8;
    int local[8];
    int sum = 0;
    for (int i = 0; i < 8; ++i) {
        local[i] = sum;
        sum += (base + i < nb) ? cnt[base + i] : 0;
    }
    s[t] = sum;
    __syncthreads();
    for (int d = 1; d < 256; d <<= 1) {
        int v = (t >= d) ? s[t - d] : 0;
        __syncthreads();
        if (t >= d) s[t] += v;
        __syncthreads();
    }
    const int prev = (t == 0) ? 0 : s[t - 1];
    for (int i = 0; i < 8; ++i)
        if (base + i < nb) off[base + i] = prev + local[i];
    if (t == 255) off[nb] = s[255];   // total valid count K
}

// ---------------- Pass C: stable compaction, writes every output slot ----------------
__global__ void nl_compact_kernel(const float* __restrict__ coords,
                                  const int*   __restrict__ nl,
                                  const float* __restrict__ sv,
                                  const int*   __restrict__ bOff,
                                  float* __restrict__ out) {
    __shared__ float sC[NATOMS * 3];
    __shared__ int   sWaveCnt[8];
    __shared__ int   sWaveOff[8];
    __shared__ int   sRun;
    const int mol = blockIdx.y;
    const int pb  = blockIdx.x;

    if (threadIdx.x == 0) sRun = bOff[mol * PB + pb];  // global exclusive base
    stage_coords(sC, coords, mol);                     // ends with __syncthreads()

    const int K = bOff[NB];
    float* out_nl0 = out;
    float* out_nl1 = out + TOTAL;
    float* out_sv  = out + 2 * TOTAL;

    const int lane = threadIdx.x & 31;
    const int wave = threadIdx.x >> 5;
    const long long jbase = (long long)mol * NPAIRS;

    for (int it = 0; it < ITERS; ++it) {
        const int pair = pb * CHUNK + it * BLOCK + threadIdx.x;
        const bool act = pair < NPAIRS;
        int a = 0, b = 0;
        float s0 = 0.f, s1 = 0.f, s2 = 0.f;
        bool valid = false;
        if (act) {
            a  = nl[pair];
            b  = nl[NPAIRS + pair];
            s0 = sv[pair * 3 + 0];
            s1 = sv[pair * 3 + 1];
            s2 = sv[pair * 3 + 2];
            const float dx = sC[a * 3 + 0] - sC[b * 3 + 0] + s0;
            const float dy = sC[a * 3 + 1] - sC[b * 3 + 1] + s1;
            const float dz = sC[a * 3 + 2] - sC[b * 3 + 2] + s2;
            valid = (dx * dx + dy * dy + dz * dz) <= CUT2;
        }
        const unsigned mb = __builtin_amdgcn_ballot_w32(valid);
        if (lane == 0) sWaveCnt[wave] = __popc(mb);
        __syncthreads();
        if (threadIdx.x == 0) {
            int acc = sRun;
            for (int w = 0; w < 8; ++w) { sWaveOff[w] = acc; acc += sWaveCnt[w]; }
            sRun = acc;
        }
        __syncthreads();
        if (act) {
            // # of valid items strictly before this item, globally (stable order)
            const int before = sWaveOff[wave] + __popc(mb & ((1u << lane) - 1u));
            if (valid) {
                const long long pos = before;
                out_nl0[pos] = (float)(a + mol * NATOMS);
                out_nl1[pos] = (float)(b + mol * NATOMS);
                out_sv[pos * 3 + 0] = s0;
                out_sv[pos * 3 + 1] = s1;
                out_sv[pos * 3 + 2] = s2;
            } else {
                const long long j   = jbase + pair;
                const long long pos = (long long)K + (j - before);
                out_nl0[pos] = -1.0f;
                out_nl1[pos] = -1.0f;
                out_sv[pos * 3 + 0] = 0.f;
                out_sv[pos * 3 + 1] = 0.f;
                out_sv[pos * 3 + 2] = 0.f;
            }
        }
    }
}

extern "C" void kernel_launch(void* const* d_in, const int* in_sizes, int n_in,
                              void* d_out, int out_size, void* d_ws, size_t ws_size,
                              hipStream_t stream) {
    (void)in_sizes; (void)n_in; (void)out_size; (void)ws_size;
    const float* coords = (const float*)d_in[0];
    const int*   nl     = (const int*)d_in[1];
    const float* sv     = (const float*)d_in[2];
    float*       out    = (float*)d_out;

    int* blkCnt = (int*)d_ws;        // NB counts
    int* blkOff = blkCnt + NB;       // NB+1 exclusive offsets (+ total K)

    dim3 grid(PB, NMOLS);
    nl_count_kernel<<<grid, BLOCK, 0, stream>>>(coords, nl, sv, blkCnt);
    nl_scan_kernel<<<1, 256, 0, stream>>>(blkCnt, blkOff, NB);
    nl_compact_kernel<<<grid, BLOCK, 0, stream>>>(coords, nl, sv, blkOff, out);
}